// GCN_46583215292438
// MI455X (gfx1250) — compile-verified
//
#include <hip/hip_runtime.h>
#include <hip/hip_bf16.h>

#define N_NODES      50000
#define N_EDGES      800000
#define IN_FEATS     128
#define HIDDEN       96
#define NUM_CLASSES  21
#define LEAKY_SLOPE  0.01f

typedef float v2f __attribute__((ext_vector_type(2)));
typedef float v8f __attribute__((ext_vector_type(8)));

// ---------------------------------------------------------------------------
// Degree / normalization: deg[i] = 1 (self loop) + in-degree; dinv = rsqrt(deg)
// ---------------------------------------------------------------------------
__global__ void init_deg_kernel(float* __restrict__ deg) {
    int i = blockIdx.x * blockDim.x + threadIdx.x;
    if (i < N_NODES) deg[i] = 1.0f;
}

__global__ void deg_scatter_kernel(const int* __restrict__ ei, float* __restrict__ deg) {
    int e = blockIdx.x * blockDim.x + threadIdx.x;
    if (e < N_EDGES) {
        int tgt = ei[N_EDGES + e];        // edge_index[1][e]
        atomicAdd(&deg[tgt], 1.0f);
    }
}

__global__ void finalize_dinv_kernel(float* __restrict__ dinv) {
    int i = blockIdx.x * blockDim.x + threadIdx.x;
    if (i < N_NODES) {
        float d = dinv[i];
        dinv[i] = (d > 0.0f) ? rsqrtf(d) : 0.0f;
    }
}

// ---------------------------------------------------------------------------
// Fused dense transform + self-loop init, fp32 WMMA (V_WMMA_F32_16X16X4_F32):
//   T[N x M]   = A[N x K] * W[K x M]
//   AGG[N x M] = dinv^2 * T          (self-loop term of the aggregation)
//
// W is staged once per block into LDS (<= 48 KB of 320 KB/WGP); each wave32
// owns one 16-row stripe and ALL M/16 column tiles (independent accumulators
// -> no WMMA D->A/B hazards between consecutive v_wmma ops; A loaded once
// per k-step instead of per column tile).
//
// VGPR layouts per ISA 7.12.2 (f32):
//   A 16x4:  lane m = lane%16, VGPR r + lanegroup g -> K = 2g + r
//   B 4x16:  K = 2g + r, N = lane%16
//   C/D:     VGPR v, group g -> M = v + 8g, N = lane%16
// ---------------------------------------------------------------------------
template <int K, int M>
__global__ void gemm_wmma_fused_kernel(const float* __restrict__ A,
                                       const float* __restrict__ W,
                                       const float* __restrict__ dinv,
                                       float* __restrict__ T,
                                       float* __restrict__ AGG) {
    constexpr int ROW_TILES = N_NODES / 16;   // 50000 = 16 * 3125 exactly
    constexpr int COL_TILES = M / 16;

    __shared__ float sW[K * M];
    for (int i = threadIdx.x; i < K * M; i += blockDim.x)
        sW[i] = W[i];
    __syncthreads();

    int wave = blockIdx.x * (blockDim.x >> 5) + (threadIdx.x >> 5);
    if (wave >= ROW_TILES) return;            // wave-uniform: EXEC stays all-1s

    int lane = threadIdx.x & 31;
    int g    = lane >> 4;                     // lane group
    int t16  = lane & 15;
    int row  = wave * 16 + t16;               // A-matrix row for this lane

    v8f acc[COL_TILES];
#pragma unroll
    for (int c = 0; c < COL_TILES; ++c) acc[c] = (v8f){};

    for (int k0 = 0; k0 < K; k0 += 4) {
        int ka = k0 + 2 * g;                              // K = 2g + r
        v2f a = *(const v2f*)(A + row * K + ka);          // 8B-aligned (ka even)
#pragma unroll
        for (int c = 0; c < COL_TILES; ++c) {
            int col = c * 16 + t16;
            v2f b;
            b.x = sW[(ka + 0) * M + col];
            b.y = sW[(ka + 1) * M + col];
            acc[c] = __builtin_amdgcn_wmma_f32_16x16x4_f32(
                false, a, false, b, (short)0, acc[c], false, false);
        }
    }

#pragma unroll
    for (int v = 0; v < 8; ++v) {
        int r = wave * 16 + v + 8 * g;                    // D: M = v + 8g
        float d  = dinv[r];
        float d2 = d * d;
#pragma unroll
        for (int c = 0; c < COL_TILES; ++c) {
            float val = acc[c][v];
            T[r * M + c * 16 + t16]   = val;
            AGG[r * M + c * 16 + t16] = d2 * val;
        }
    }
}

// ---------------------------------------------------------------------------
// Edge scatter: agg[tgt][f..f+3] += dinv[src]*dinv[tgt] * t[src][f..f+3]
// float4 per thread: 128-bit coalesced gathers, 1/4 the index traffic; the
// 19 MB feature matrix is L2-resident on MI455X (192 MB L2), so both the
// gathers and the f32 atomics stay on-chip.
// ---------------------------------------------------------------------------
#define FEAT_VECS (HIDDEN / 4)   // 24 float4 per edge

__global__ void edge_scatter_kernel(const int* __restrict__ ei,
                                    const float* __restrict__ t,
                                    const float* __restrict__ dinv,
                                    float* __restrict__ agg) {
    int idx = blockIdx.x * blockDim.x + threadIdx.x;
    if (idx >= N_EDGES * FEAT_VECS) return;
    int e = idx / FEAT_VECS;
    int q = idx - e * FEAT_VECS;
    int f = q * 4;
    int s  = ei[e];              // edge_index[0][e] (source)
    int tg = ei[N_EDGES + e];    // edge_index[1][e] (target / aggregation idx)
    float nrm = dinv[s] * dinv[tg];
    const float4 tv = *(const float4*)(t + s * HIDDEN + f);   // 16B-aligned
    float* dst = agg + tg * HIDDEN + f;
    atomicAdd(dst + 0, nrm * tv.x);
    atomicAdd(dst + 1, nrm * tv.y);
    atomicAdd(dst + 2, nrm * tv.z);
    atomicAdd(dst + 3, nrm * tv.w);
}

// ---------------------------------------------------------------------------
// h[i][f] = leaky_relu(agg[i][f] + b[f])  (float4 vectorized)
// ---------------------------------------------------------------------------
__global__ void bias_lrelu_kernel(const float* __restrict__ agg,
                                  const float* __restrict__ b,
                                  float* __restrict__ h) {
    int idx = (blockIdx.x * blockDim.x + threadIdx.x) * 4;
    if (idx >= N_NODES * HIDDEN) return;
    int f = idx % HIDDEN;
    float4 x  = *(const float4*)(agg + idx);
    float4 bb = *(const float4*)(b + f);
    float4 r;
    r.x = x.x + bb.x; r.x = (r.x > 0.0f) ? r.x : LEAKY_SLOPE * r.x;
    r.y = x.y + bb.y; r.y = (r.y > 0.0f) ? r.y : LEAKY_SLOPE * r.y;
    r.z = x.z + bb.z; r.z = (r.z > 0.0f) ? r.z : LEAKY_SLOPE * r.z;
    r.w = x.w + bb.w; r.w = (r.w > 0.0f) ? r.w : LEAKY_SLOPE * r.w;
    *(float4*)(h + idx) = r;
}

// ---------------------------------------------------------------------------
// Head: out[N x 21] = h[N x 96] * Wl[96 x 21] + bl.
// Wl is staged zero-padded to 96x32 in LDS -> unconditional B loads (no
// divergent branch ladders around the WMMA loop); store masked to 21 cols.
// ---------------------------------------------------------------------------
__global__ void gemm_head_kernel(const float* __restrict__ A,
                                 const float* __restrict__ W,
                                 const float* __restrict__ bias,
                                 float* __restrict__ out) {
    constexpr int K  = HIDDEN;
    constexpr int M  = NUM_CLASSES;
    constexpr int MP = 32;                    // padded columns (2 tiles)
    constexpr int ROW_TILES = N_NODES / 16;

    __shared__ float sW[K * MP];
    for (int i = threadIdx.x; i < K * MP; i += blockDim.x) {
        int k = i / MP, c = i - k * MP;
        sW[i] = (c < M) ? W[k * M + c] : 0.0f;
    }
    __syncthreads();

    int wave = blockIdx.x * (blockDim.x >> 5) + (threadIdx.x >> 5);
    if (wave >= ROW_TILES) return;

    int lane = threadIdx.x & 31;
    int g    = lane >> 4;
    int t16  = lane & 15;
    int row  = wave * 16 + t16;

    v8f acc[2];
    acc[0] = (v8f){};
    acc[1] = (v8f){};

#pragma unroll
    for (int k0 = 0; k0 < K; k0 += 4) {
        int ka = k0 + 2 * g;
        v2f a = *(const v2f*)(A + row * K + ka);
#pragma unroll
        for (int c = 0; c < 2; ++c) {
            int col = c * 16 + t16;
            v2f b;
            b.x = sW[(ka + 0) * MP + col];
            b.y = sW[(ka + 1) * MP + col];
            acc[c] = __builtin_amdgcn_wmma_f32_16x16x4_f32(
                false, a, false, b, (short)0, acc[c], false, false);
        }
    }

#pragma unroll
    for (int v = 0; v < 8; ++v) {
        int r = wave * 16 + v + 8 * g;
#pragma unroll
        for (int c = 0; c < 2; ++c) {
            int col = c * 16 + t16;
            if (col < M)                                   // post-WMMA divergence: safe
                out[r * M + col] = acc[c][v] + bias[col];
        }
    }
}

// ---------------------------------------------------------------------------
extern "C" void kernel_launch(void* const* d_in, const int* in_sizes, int n_in,
                              void* d_out, int out_size, void* d_ws, size_t ws_size,
                              hipStream_t stream) {
    (void)in_sizes; (void)n_in; (void)out_size; (void)ws_size;

    const float* x  = (const float*)d_in[0];
    const int*   ei = (const int*)d_in[1];          // [2, E] int32 (JAX x64 off)
    const float* W1 = (const float*)d_in[2];
    const float* b1 = (const float*)d_in[3];
    const float* W2 = (const float*)d_in[4];
    const float* b2 = (const float*)d_in[5];
    const float* W3 = (const float*)d_in[6];
    const float* b3 = (const float*)d_in[7];
    const float* Wl = (const float*)d_in[8];
    const float* bl = (const float*)d_in[9];
    float* out = (float*)d_out;

    // Workspace layout (~58 MB): dinv | t | agg | h
    char* ws = (char*)d_ws;
    size_t off = 0;
    float* dinv = (float*)(ws + off); off += ((size_t)N_NODES * 4 + 255) & ~(size_t)255;
    float* tbuf = (float*)(ws + off); off += (size_t)N_NODES * HIDDEN * 4;
    float* agg  = (float*)(ws + off); off += (size_t)N_NODES * HIDDEN * 4;
    float* hbuf = (float*)(ws + off);

    const dim3 blk(256);
    const int nodeBlocks = (N_NODES + 255) / 256;
    const int edgeBlocks = (N_EDGES + 255) / 256;
    const int reluBlocks = (N_NODES * HIDDEN / 4 + 255) / 256;            // 4688
    const int scatBlocks = (N_EDGES * FEAT_VECS + 255) / 256;             // 75000
    const int gemmBlocks = ((N_NODES / 16) + 7) / 8;                      // 391 (8 waves/block)

    // Normalization (shared by all layers)
    init_deg_kernel<<<nodeBlocks, blk, 0, stream>>>(dinv);
    deg_scatter_kernel<<<edgeBlocks, blk, 0, stream>>>(ei, dinv);
    finalize_dinv_kernel<<<nodeBlocks, blk, 0, stream>>>(dinv);

    // Layer 1: x[N,128] -> h[N,96]
    gemm_wmma_fused_kernel<IN_FEATS, HIDDEN><<<gemmBlocks, blk, 0, stream>>>(x, W1, dinv, tbuf, agg);
    edge_scatter_kernel<<<scatBlocks, blk, 0, stream>>>(ei, tbuf, dinv, agg);
    bias_lrelu_kernel<<<reluBlocks, blk, 0, stream>>>(agg, b1, hbuf);

    // Layer 2
    gemm_wmma_fused_kernel<HIDDEN, HIDDEN><<<gemmBlocks, blk, 0, stream>>>(hbuf, W2, dinv, tbuf, agg);
    edge_scatter_kernel<<<scatBlocks, blk, 0, stream>>>(ei, tbuf, dinv, agg);
    bias_lrelu_kernel<<<reluBlocks, blk, 0, stream>>>(agg, b2, hbuf);

    // Layer 3
    gemm_wmma_fused_kernel<HIDDEN, HIDDEN><<<gemmBlocks, blk, 0, stream>>>(hbuf, W3, dinv, tbuf, agg);
    edge_scatter_kernel<<<scatBlocks, blk, 0, stream>>>(ei, tbuf, dinv, agg);
    bias_lrelu_kernel<<<reluBlocks, blk, 0, stream>>>(agg, b3, hbuf);

    // Head: h[N,96] @ Wl[96,21] + bl
    gemm_head_kernel<<<gemmBlocks, blk, 0, stream>>>(hbuf, Wl, bl, out);
}